// PolyConv_72043781423165
// MI455X (gfx1250) — compile-verified
//
#include <hip/hip_runtime.h>
#include <stdint.h>

#define DFEAT       64
#define TILE_EDGES  1024
#define NT          256

// ---------------- CDNA5 async global->LDS helpers ----------------
typedef int v4i __attribute__((vector_size(16)));
typedef __attribute__((address_space(1))) v4i* as1_v4i_p;   // global
typedef __attribute__((address_space(3))) v4i* as3_v4i_p;   // LDS

__device__ __forceinline__ void async_copy_b128(const void* gsrc, void* ldst) {
#if __has_builtin(__builtin_amdgcn_global_load_async_to_lds_b128)
  __builtin_amdgcn_global_load_async_to_lds_b128(
      (as1_v4i_p)(uintptr_t)gsrc,
      (as3_v4i_p)(uint32_t)(uintptr_t)ldst,
      0, 0);
#else
  uint32_t loff = (uint32_t)(uintptr_t)ldst;   // low 32 bits of generic LDS ptr = LDS offset
  asm volatile("global_load_async_to_lds_b128 %0, %1, off"
               :: "v"(loff), "v"(gsrc) : "memory");
#endif
}

__device__ __forceinline__ void wait_async_zero() {
#if __has_builtin(__builtin_amdgcn_s_wait_asynccnt)
  __builtin_amdgcn_s_wait_asynccnt(0);
#else
  asm volatile("s_wait_asynccnt 0" ::: "memory");
#endif
}

__device__ __forceinline__ void atomAddF(float* p, float v) {
  __hip_atomic_fetch_add(p, v, __ATOMIC_RELAXED, __HIP_MEMORY_SCOPE_AGENT);
}

// ---------------- elementwise kernels ----------------
__global__ void zero_f32_k(float* __restrict__ p, int n) {
  int i = blockIdx.x * blockDim.x + threadIdx.x;
  if (i < n) p[i] = 0.0f;
}

__global__ void zero_f4_k(float4* __restrict__ p, int n4) {
  int i = blockIdx.x * blockDim.x + threadIdx.x;
  if (i < n4) p[i] = make_float4(0.f, 0.f, 0.f, 0.f);
}

__global__ void degree_k(const int* __restrict__ col, float* __restrict__ deg, int E) {
  int e = blockIdx.x * blockDim.x + threadIdx.x;
  if (e < E) atomAddF(&deg[col[e]], 1.0f);
}

__global__ void dinv_k(float* __restrict__ deg, int N) {
  int i = blockIdx.x * blockDim.x + threadIdx.x;
  if (i < N) {
    float d = deg[i];
    deg[i] = (d > 0.0f) ? rsqrtf(d) : 0.0f;   // deg is integer-valued, max(deg,1)==deg when >0
  }
}

// compress indices to tile-padded int32 arrays + per-edge weight
__global__ void prep_k(const int* __restrict__ idx, const float* __restrict__ dinv,
                       int* __restrict__ row32, int* __restrict__ col32,
                       float* __restrict__ ew, int E, int Epad) {
  int e = blockIdx.x * blockDim.x + threadIdx.x;
  if (e < E) {
    int r = idx[e];
    int c = idx[E + e];
    row32[e] = r;
    col32[e] = c;
    ew[e]    = dinv[r] * dinv[c];
  } else if (e < Epad) {          // defined padding so async tile staging never reads garbage
    row32[e] = 0;
    col32[e] = 0;
    ew[e]    = 0.0f;
  }
}

// ---------------- scatter SpMM: out[col] += ew * feat[row] ----------------
// Block = 256 threads = 8 waves; tile = 1024 edges staged to LDS asynchronously
// (global_load_async_to_lds_b128 + s_wait_asynccnt). Each wave owns 128 edges;
// a half-wave (16 lanes) handles one edge: one b128 gather per lane (64
// floats/edge) + 4 native f32 atomics into the target row. All hot data
// (12.8 MB features + 12.8 MB accumulator + edge stream) is L2-resident on
// MI455X (192 MB L2), so this is L2-atomic-throughput bound by construction.
__global__ __launch_bounds__(NT) void scatter_k(const int*   __restrict__ row,
                                                const int*   __restrict__ col,
                                                const float* __restrict__ ew,
                                                const float* __restrict__ feat,
                                                float*       __restrict__ out,
                                                int E) {
  __shared__ int   rl[TILE_EDGES];
  __shared__ int   cl[TILE_EDGES];
  __shared__ float wl[TILE_EDGES];

  const int t        = threadIdx.x;
  const int tileBase = blockIdx.x * TILE_EDGES;

  // Stage this block's edge parameters: 16 B per thread per array (3 async ops/thread)
  async_copy_b128(row + tileBase + t * 4, &rl[t * 4]);
  async_copy_b128(col + tileBase + t * 4, &cl[t * 4]);
  async_copy_b128(ew  + tileBase + t * 4, &wl[t * 4]);
  wait_async_zero();
  __syncthreads();

  const int wave = t >> 5;
  const int half = (t >> 4) & 1;     // which edge of the current pair
  const int sub  = t & 15;           // 16 lanes x float4 = 64 features
  const int base = wave * (TILE_EDGES / 8);   // 128 edges per wave

  #pragma unroll 4
  for (int j = 0; j < TILE_EDGES / 8; j += 2) {
    const int el = base + j + half;
    const int e  = tileBase + el;
    if (e < E) {
      const int   r = rl[el];
      const int   c = cl[el];
      const float w = wl[el];
      const float4 v = *(const float4*)(feat + (r << 6) + (sub << 2));
      float* o = out + (c << 6) + (sub << 2);
      atomAddF(o + 0, w * v.x);
      atomAddF(o + 1, w * v.y);
      atomAddF(o + 2, w * v.z);
      atomAddF(o + 3, w * v.w);
    }
  }
}

// ---------------- fused updates ----------------
// Each update pass also re-zeroes `out` for the next scatter round, saving a
// separate 12.8 MB zeroing dispatch per iteration.
// k==1:  feat = x - out;  h = th0*x + th1*feat;  out = 0
__global__ void update_first_k(const float4* __restrict__ x, float4* __restrict__ out,
                               float4* __restrict__ feat, float4* __restrict__ h,
                               int n4, float th0, float th1) {
  int i = blockIdx.x * blockDim.x + threadIdx.x;
  if (i < n4) {
    float4 xv = x[i], ov = out[i], f, hv;
    f.x = xv.x - ov.x; f.y = xv.y - ov.y; f.z = xv.z - ov.z; f.w = xv.w - ov.w;
    feat[i] = f;
    hv.x = th0 * xv.x + th1 * f.x;
    hv.y = th0 * xv.y + th1 * f.y;
    hv.z = th0 * xv.z + th1 * f.z;
    hv.w = th0 * xv.w + th1 * f.w;
    h[i] = hv;
    out[i] = make_float4(0.f, 0.f, 0.f, 0.f);
  }
}

// k>=2:  feat -= out;  h += th*feat;  out = 0
__global__ void update_k(float4* __restrict__ out, float4* __restrict__ feat,
                         float4* __restrict__ h, int n4, float th) {
  int i = blockIdx.x * blockDim.x + threadIdx.x;
  if (i < n4) {
    float4 f = feat[i], o = out[i], hv = h[i];
    f.x -= o.x; f.y -= o.y; f.z -= o.z; f.w -= o.w;
    feat[i] = f;
    hv.x += th * f.x; hv.y += th * f.y; hv.z += th * f.z; hv.w += th * f.w;
    h[i] = hv;
    out[i] = make_float4(0.f, 0.f, 0.f, 0.f);
  }
}

// ---------------- host orchestration ----------------
extern "C" void kernel_launch(void* const* d_in, const int* in_sizes, int n_in,
                              void* d_out, int out_size, void* d_ws, size_t ws_size,
                              hipStream_t stream) {
  const float* x   = (const float*)d_in[0];
  const int*   idx = (const int*)d_in[1];          // [2, E] flattened: row then col

  const int N      = in_sizes[0] / DFEAT;
  const int E      = in_sizes[1] / 2;
  const int nTiles = (E + TILE_EDGES - 1) / TILE_EDGES;
  const int Epad   = nTiles * TILE_EDGES;
  const int NF     = N * DFEAT;
  const int n4     = NF / 4;

  auto align256 = [](size_t v) { return (v + 255) & ~(size_t)255; };
  char* p = (char*)d_ws;
  float* deg   = (float*)p;  p += align256((size_t)N * 4);      // degree, then dinv in place
  int*   row32 = (int*)p;    p += align256((size_t)Epad * 4);
  int*   col32 = (int*)p;    p += align256((size_t)Epad * 4);
  float* ew    = (float*)p;  p += align256((size_t)Epad * 4);
  float* feat  = (float*)p;  p += align256((size_t)NF * 4);
  float* outb  = (float*)p;  p += align256((size_t)NF * 4);
  float* h     = (float*)d_out;

  // normalization precompute (outb zeroed concurrently-serial on same stream)
  zero_f32_k<<<(N + NT - 1) / NT, NT, 0, stream>>>(deg, N);
  degree_k  <<<(E + NT - 1) / NT, NT, 0, stream>>>(idx + E, deg, E);
  dinv_k    <<<(N + NT - 1) / NT, NT, 0, stream>>>(deg, N);
  prep_k    <<<(Epad + NT - 1) / NT, NT, 0, stream>>>(idx, deg, row32, col32, ew, E, Epad);
  zero_f4_k <<<(n4 + NT - 1) / NT, NT, 0, stream>>>((float4*)outb, n4);

  const float theta[5] = {1.0f, 0.5f, 0.25f, 0.125f, 0.0625f};
  for (int k = 1; k <= 4; ++k) {
    scatter_k<<<nTiles, NT, 0, stream>>>(row32, col32, ew,
                                         (k == 1) ? x : feat, outb, E);
    if (k == 1)
      update_first_k<<<(n4 + NT - 1) / NT, NT, 0, stream>>>(
          (const float4*)x, (float4*)outb, (float4*)feat, (float4*)h,
          n4, theta[0], theta[1]);
    else
      update_k<<<(n4 + NT - 1) / NT, NT, 0, stream>>>(
          (float4*)outb, (float4*)feat, (float4*)h, n4, theta[k]);
  }
}